// LightGCNStack_65223373357476
// MI455X (gfx1250) — compile-verified
//
#include <hip/hip_runtime.h>

#define NUM_USERS   100000
#define NUM_ARTISTS 50000
#define NUM_LAYERS  3
#define NUM_EDGES   1000000
#define DIM         64

typedef __attribute__((ext_vector_type(2))) float v2f;
typedef __attribute__((ext_vector_type(8))) float v8f;

// ---------------------------------------------------------------------------
// Hardware FP32 atomic add (global_atomic_add_f32, no CAS loop).
// ---------------------------------------------------------------------------
__device__ __forceinline__ void atomAddF(float* p, float v) {
    unsafeAtomicAdd(p, v);
}

// ---------------------------------------------------------------------------
// Degree counting: cnt_a[src]++ , cnt_u[dst]++  (counts as exact float ints)
// ---------------------------------------------------------------------------
__global__ __launch_bounds__(256)
void count_kernel(const int* __restrict__ e_src_a,
                  const int* __restrict__ e_dst_u,
                  float* __restrict__ cnt_u,
                  float* __restrict__ cnt_a,
                  int n_edges) {
    int e = blockIdx.x * blockDim.x + threadIdx.x;
    if (e < n_edges) {
        atomAddF(&cnt_a[e_src_a[e]], 1.0f);
        atomAddF(&cnt_u[e_dst_u[e]], 1.0f);
    }
}

// In-place: c -> (c > 0 ? 1/c : 0)   (matches scatter-mean empty-segment rule)
__global__ __launch_bounds__(256)
void invert_kernel(float* __restrict__ v, int n) {
    int i = blockIdx.x * blockDim.x + threadIdx.x;
    if (i < n) {
        float c = v[i];
        v[i] = (c > 0.0f) ? (1.0f / c) : 0.0f;
    }
}

// ---------------------------------------------------------------------------
// fin_u = x_users*scale ; fin_a = x_artists*scale  (float4 vectorized, exact grid)
// ---------------------------------------------------------------------------
__global__ __launch_bounds__(256)
void init_fin_kernel(const float4* __restrict__ xu,
                     const float4* __restrict__ xa,
                     float4* __restrict__ fin,   // users block then artists block
                     float scale) {
    const int NU4 = NUM_USERS * (DIM / 4);
    int t = blockIdx.x * blockDim.x + threadIdx.x;   // exact: (NU+NA)*16 threads
    float4 s = (t < NU4) ? xu[t] : xa[t - NU4];
    float4 o;
    o.x = s.x * scale; o.y = s.y * scale; o.z = s.z * scale; o.w = s.w * scale;
    fin[t] = o;
}

// ---------------------------------------------------------------------------
// Edge scatter-add: acc[s_idx[e]] += feat[g_idx[e]]  (64 dims / edge)
// 16 threads per edge, each thread: one float4 gather + 4 HW f32 atomics.
// Tables are L2-resident (192 MB L2 >> 38 MB working set) -> atomics in L2.
// Exact grid (NUM_EDGES*16 threads), no branches, full coalescing per row.
// ---------------------------------------------------------------------------
__global__ __launch_bounds__(256)
void scatter_kernel(const float* __restrict__ feat,
                    const int*   __restrict__ g_idx,
                    const int*   __restrict__ s_idx,
                    float*       __restrict__ acc) {
    int t = blockIdx.x * blockDim.x + threadIdx.x;
    int e = t >> 4;           // edge id
    int q = t & 15;           // which float4 of the 64-dim row
    int gs = g_idx[e];
    int sd = s_idx[e];
    const float4* srow = reinterpret_cast<const float4*>(feat + (size_t)gs * DIM);
    float4 v = srow[q];
    float* drow = acc + (size_t)sd * DIM + (q << 2);
    atomAddF(drow + 0, v.x);
    atomAddF(drow + 1, v.y);
    atomAddF(drow + 2, v.z);
    atomAddF(drow + 3, v.w);
}

// ---------------------------------------------------------------------------
// Finalize via WMMA:  X = diag(inv) x ACC  (per 16x16 tile, 4 chained
// V_WMMA_F32_16X16X4_F32), then ACC <- X (in place) and FIN += X*scale.
//
// Block = 128 threads = 4 waves; wave w owns column tile w of a 16-row block.
// Row counts (100000, 50000) are multiples of 16 and DIM=64=4*16, so every
// wave is full -> EXEC all-ones as WMMA requires; no divergence anywhere.
//
// Layouts per ISA 7.12.2 (32-bit, wave32):
//   A 16x4 : lane m=lane%16 holds K=(lane<16?{0,1}:{2,3}) in {a.x,a.y}
//   B 4x16 : lane n=lane%16 holds K=(lane<16?{0,1}:{2,3}) rows in {b.x,b.y}
//   C 16x16: c[v] is element (M = v + (lane>=16 ? 8 : 0), N = lane%16)
// ---------------------------------------------------------------------------
__global__ __launch_bounds__(128)
void finalize_kernel(float* __restrict__ acc,        // in: sums, out: means
                     const float* __restrict__ inv,  // 1/degree (0 if empty)
                     float* __restrict__ fin,        // running average accum
                     float scale) {
    const int wave    = threadIdx.x >> 5;        // 0..3 -> column tile
    const int lane    = threadIdx.x & 31;
    const int rowbase = blockIdx.x << 4;         // 16 rows per block
    const int colbase = wave << 4;               // 16 cols per wave
    const int m       = lane & 15;               // A: M index / B,C: N index
    const int khalf   = (lane >> 4) << 1;        // 0 (lanes 0-15) or 2 (16-31)

    const float invm = inv[rowbase + m];

    v8f c = {};
#pragma unroll
    for (int t = 0; t < 4; ++t) {
        const int k0 = (t << 2) + khalf;         // global diag position of a.x
        v2f a, b;
        a.x = (m == k0)     ? invm : 0.0f;       // A_t = 16x4 slice of diag(inv)
        a.y = (m == k0 + 1) ? invm : 0.0f;
        b.x = acc[(size_t)(rowbase + k0)     * DIM + colbase + m];
        b.y = acc[(size_t)(rowbase + k0 + 1) * DIM + colbase + m];
        c = __builtin_amdgcn_wmma_f32_16x16x4_f32(
                /*neg_a=*/false, a, /*neg_b=*/false, b,
                /*c_mod=*/(short)0, c, /*reuse_a=*/false, /*reuse_b=*/false);
    }

    const int mtop = (lane >> 4) << 3;           // 0 or 8
#pragma unroll
    for (int v = 0; v < 8; ++v) {
        const size_t idx = (size_t)(rowbase + mtop + v) * DIM + colbase + m;
        const float x = c[v];
        acc[idx] = x;                            // mean embedding for next hop
        fin[idx] += x * scale;                   // layer-averaged output
    }
}

// ---------------------------------------------------------------------------
// Host launcher
// ---------------------------------------------------------------------------
extern "C" void kernel_launch(void* const* d_in, const int* in_sizes, int n_in,
                              void* d_out, int out_size, void* d_ws, size_t ws_size,
                              hipStream_t stream) {
    (void)in_sizes; (void)n_in; (void)out_size; (void)ws_size;

    const float* x_users   = (const float*)d_in[0];
    const float* x_artists = (const float*)d_in[1];
    const int*   e_src_a   = (const int*)d_in[2];
    const int*   e_dst_u   = (const int*)d_in[3];

    float* fin_u = (float*)d_out;                         // 100000*64
    float* fin_a = fin_u + (size_t)NUM_USERS * DIM;       //  50000*64

    // Workspace: [xu buf][xa buf][inv_u][inv_a]  (~39 MB)
    float* xu    = (float*)d_ws;
    float* xa    = xu + (size_t)NUM_USERS * DIM;
    float* inv_u = xa + (size_t)NUM_ARTISTS * DIM;
    float* inv_a = inv_u + NUM_USERS;

    const float scale = 1.0f / (NUM_LAYERS + 1);

    // 1) reciprocal degrees (inv_u, inv_a are contiguous)
    hipMemsetAsync(inv_u, 0, (size_t)(NUM_USERS + NUM_ARTISTS) * sizeof(float), stream);
    count_kernel<<<(NUM_EDGES + 255) / 256, 256, 0, stream>>>(
        e_src_a, e_dst_u, inv_u, inv_a, NUM_EDGES);
    invert_kernel<<<(NUM_USERS + NUM_ARTISTS + 255) / 256, 256, 0, stream>>>(
        inv_u, NUM_USERS + NUM_ARTISTS);

    // 2) fin = x0 * scale   (exact grid: (NU+NA)*16 float4 threads)
    init_fin_kernel<<<(NUM_USERS + NUM_ARTISTS) * (DIM / 4) / 256, 256, 0, stream>>>(
        (const float4*)x_users, (const float4*)x_artists, (float4*)d_out, scale);

    // 3) propagation layers
    const int SCATTER_BLOCKS = NUM_EDGES * 16 / 256;      // exact
    const float* cur_a = x_artists;                       // layer 0 reads input
    for (int l = 0; l < NUM_LAYERS; ++l) {
        // a2u: users <- mean over incoming artist rows
        hipMemsetAsync(xu, 0, (size_t)NUM_USERS * DIM * sizeof(float), stream);
        scatter_kernel<<<SCATTER_BLOCKS, 256, 0, stream>>>(cur_a, e_src_a, e_dst_u, xu);
        finalize_kernel<<<NUM_USERS / 16, 128, 0, stream>>>(xu, inv_u, fin_u, scale);

        // u2a: artists <- mean over the UPDATED user rows (sequential dependency)
        hipMemsetAsync(xa, 0, (size_t)NUM_ARTISTS * DIM * sizeof(float), stream);
        scatter_kernel<<<SCATTER_BLOCKS, 256, 0, stream>>>(xu, e_dst_u, e_src_a, xa);
        finalize_kernel<<<NUM_ARTISTS / 16, 128, 0, stream>>>(xa, inv_a, fin_a, scale);

        cur_a = xa;
    }
}